// CoxPHLossStratified_53566832115884
// MI455X (gfx1250) — compile-verified
//
#include <hip/hip_runtime.h>
#include <hip/hip_bf16.h>

#define NBINS   8192
#define KSTRATA 64
#define TOTB    (KSTRATA * NBINS)
#define EPSF    1e-7f
#define MAXB    512
#define WPB     8                 // waves per block in main kernel
#define PF_BLOCK (WPB * 32)

typedef float v2f __attribute__((ext_vector_type(2)));
typedef float v8f __attribute__((ext_vector_type(8)));
typedef unsigned long long ull;

// ---- monotone float<->uint order mapping (handles negatives, for log_h max) ----
__device__ __forceinline__ unsigned monoKey(float f) {
    unsigned u = __float_as_uint(f);
    return (u & 0x80000000u) ? ~u : (u | 0x80000000u);
}
__device__ __forceinline__ float unmono(unsigned u) {
    unsigned b = (u & 0x80000000u) ? (u & 0x7FFFFFFFu) : ~u;
    return __uint_as_float(b);
}
// durations are >= 0: raw bits are already order-monotone
__device__ __forceinline__ int binOf(float d) {
    int b = (int)(d * (float)(NBINS / 100.0));   // 81.92f, monotone in d
    if (b < 0) b = 0;
    if (b >= NBINS) b = NBINS - 1;
    return b;
}
// composite order key: c_j >= c_i  <=>  d_j > d_i, or d_j == d_i and idx_j <= idx_i.
// (indices are unique, so cross-element ties are impossible; self-compare -> 1.)
// ev rides in bit 0 and never affects ordering between distinct elements.
__device__ __forceinline__ ull compKey(unsigned durKey, unsigned idxev) {
    unsigned id = idxev & 0x7FFFFFFFu;
    return ((ull)durKey << 32) | (((ull)(0x7FFFFFFFu - id)) << 1) | (ull)(idxev >> 31);
}

// ---- pass 0: zero counts + gamma ----
__global__ void k_init(unsigned* counts, unsigned* gammaU) {
    int i = blockIdx.x * blockDim.x + threadIdx.x;
    if (i < TOTB) counts[i] = 0u;
    if (i < KSTRATA) gammaU[i] = 0u;     // == -inf under monoKey order
}

// ---- pass A: per-stratum gamma (max log_h) + bucket counts ----
__global__ void k_hist(const float* __restrict__ lh, const float* __restrict__ dur,
                       const int* __restrict__ bidx,
                       unsigned* counts, unsigned* gammaU, int n) {
    __shared__ unsigned gmax[KSTRATA];
    for (int t = threadIdx.x; t < KSTRATA; t += blockDim.x) gmax[t] = 0u;
    __syncthreads();
    int stride = gridDim.x * blockDim.x;
    for (int i = blockIdx.x * blockDim.x + threadIdx.x; i < n; i += stride) {
        int k = bidx[i];
        atomicMax(&gmax[k], monoKey(lh[i]));
        atomicAdd(&counts[k * NBINS + binOf(dur[i])], 1u);
    }
    __syncthreads();
    for (int t = threadIdx.x; t < KSTRATA; t += blockDim.x) atomicMax(&gammaU[t], gmax[t]);
}

// ---- pass B: global exclusive scan of counts -> offsets (+cursor copy) ----
__global__ void k_scanF(const unsigned* __restrict__ counts, unsigned* offsets, unsigned* cursor) {
    __shared__ unsigned part[1024];
    int t = threadIdx.x;
    const int CH = TOTB / 1024;   // 512
    unsigned s = 0u;
    for (int i = 0; i < CH; ++i) s += counts[t * CH + i];
    part[t] = s;
    __syncthreads();
    if (t == 0) {
        unsigned run = 0u;
        for (int i = 0; i < 1024; ++i) { unsigned v = part[i]; part[i] = run; run += v; }
    }
    __syncthreads();
    unsigned run = part[t];
    for (int i = 0; i < CH; ++i) {
        int idx = t * CH + i;
        unsigned v = counts[idx];
        offsets[idx] = run; cursor[idx] = run;
        run += v;
    }
}

// ---- pass C: scatter elements into bucket-sorted layout ----
__global__ void k_scatter(const float* __restrict__ lh, const float* __restrict__ dur,
                          const float* __restrict__ ev, const int* __restrict__ bidx,
                          const unsigned* __restrict__ gammaU, unsigned* cursor,
                          unsigned* skey, unsigned* sidxev, float* slhg, int n) {
    int i = blockIdx.x * blockDim.x + threadIdx.x;
    if (i >= n) return;
    int k = bidx[i];
    float d = dur[i];
    unsigned p = atomicAdd(&cursor[k * NBINS + binOf(d)], 1u);
    float gamma = unmono(gammaU[k]);
    skey[p]   = __float_as_uint(d);
    sidxev[p] = (unsigned)i | ((ev[i] > 0.5f) ? 0x80000000u : 0u);
    slhg[p]   = lh[i] - gamma;
}

// ---- pass D: deterministic per-bucket sum of e ----
__global__ void k_bsum(const unsigned* __restrict__ counts, const unsigned* __restrict__ offsets,
                       const float* __restrict__ slhg, float* bsum) {
    int b = blockIdx.x * blockDim.x + threadIdx.x;
    if (b >= TOTB) return;
    unsigned m = counts[b], base = offsets[b];
    float s = 0.f;
    for (unsigned t = 0; t < m; ++t) s += expf(slhg[base + t]);
    bsum[b] = s;
}

// ---- pass E: per-stratum reverse-exclusive scan: gsuf[bin] = sum over strictly greater bins ----
__global__ void k_scanR(const float* __restrict__ bsum, float* gsuf) {
    int k = blockIdx.x;
    const float* b = bsum + (size_t)k * NBINS;
    __shared__ float part[256];
    int t = threadIdx.x;
    const int CH = NBINS / 256;   // 32
    float s = 0.f;
    for (int i = 0; i < CH; ++i) s += b[t * CH + i];
    part[t] = s;
    __syncthreads();
    if (t == 0) {
        float run = 0.f;
        for (int i = 255; i >= 0; --i) { float v = part[i]; part[i] = run; run += v; }
    }
    __syncthreads();
    float run = part[t];
    for (int i = CH - 1; i >= 0; --i) {
        gsuf[(size_t)k * NBINS + t * CH + i] = run;
        run += b[t * CH + i];
    }
}

// ---- pass F: one wave per bucket; within-bucket masked sums via V_WMMA_F32_16X16X4_F32 ----
// Buckets are zero-padded in LDS to a multiple of 16 so the WMMA K-loop is branch-free:
// pad slots have e==0 so their mask value is irrelevant; pad rows are never consumed.
// The ordering predicate is a single u64 compare on composite keys (no short-circuit).
__global__ void __launch_bounds__(PF_BLOCK)
k_main(const unsigned* __restrict__ counts, const unsigned* __restrict__ offsets,
       const float* __restrict__ gsuf,
       const unsigned* __restrict__ skey, const unsigned* __restrict__ sidxev,
       const float* __restrict__ slhg,
       float* bnum, float* bden) {
    __shared__ __align__(16) ull   lc[WPB][MAXB];   // composite keys
    __shared__ __align__(16) float le[WPB][MAXB];   // e = exp(lh - gamma)
    int lane = threadIdx.x & 31;
    int w    = threadIdx.x >> 5;
    int wid  = blockIdx.x * WPB + w;
    int wstr = gridDim.x * WPB;
    const unsigned half2 = (lane >= 16) ? 2u : 0u;  // A/B K-split across half-waves
    const unsigned row   = (unsigned)(lane & 15);

    for (int bkt = wid; bkt < TOTB; bkt += wstr) {
        unsigned m = counts[bkt];
        if (m == 0u) { if (lane == 0) { bnum[bkt] = 0.f; bden[bkt] = 0.f; } continue; }
        unsigned base = offsets[bkt];
        float G = gsuf[bkt];
        float cnum = 0.f, cden = 0.f;

        // prefetch next bucket's scattered data (global_prefetch_b8)
        int nb = bkt + wstr;
        if (nb < TOTB) {
            unsigned nbase = offsets[nb];
            __builtin_prefetch(&skey[nbase], 0, 0);
            __builtin_prefetch(&sidxev[nbase], 0, 0);
            __builtin_prefetch(&slhg[nbase], 0, 0);
        }

        if (m <= MAXB) {
            unsigned mpad = (m + 15u) & ~15u;       // <= MAXB
            for (unsigned t = (unsigned)lane; t < m; t += 32u) {
                lc[w][t] = compKey(skey[base + t], sidxev[base + t]);
                le[w][t] = expf(slhg[base + t]);
            }
            for (unsigned t = m + (unsigned)lane; t < mpad; t += 32u) {
                lc[w][t] = 0ull;
                le[w][t] = 0.f;
            }
            asm volatile("s_wait_dscnt 0x0" ::: "memory");  // staged LDS visible wave-wide

            for (unsigned r0 = 0u; r0 < m; r0 += 16u) {
                unsigned ri = r0 + row;             // < mpad, always valid to read
                ull rc = lc[w][ri];
                v8f acc = {0.f, 0.f, 0.f, 0.f, 0.f, 0.f, 0.f, 0.f};
#pragma unroll 4
                for (unsigned j0 = 0u; j0 < mpad; j0 += 4u) {
                    unsigned jA = j0 + half2;       // even
                    ull c0 = lc[w][jA];             // consecutive pair -> ds_load_b128
                    ull c1 = lc[w][jA + 1u];
                    v2f B = *(const v2f*)&le[w][jA];        // e[jA], e[jA+1]
                    v2f A;
                    A.x = (c0 >= rc) ? 1.f : 0.f;   // single v_cmp_ge_u64 each
                    A.y = (c1 >= rc) ? 1.f : 0.f;
                    acc = __builtin_amdgcn_wmma_f32_16x16x4_f32(
                        false, A, false, B, (short)0, acc, false, false);
                }
                // All D columns equal s_m (B columns identical); pull col 0:
                // rows 0..7 live in lane 0, rows 8..15 in lane 16 (C/D layout).
                float s = 0.f;
                int src = (lane & 8) ? 16 : 0;
#pragma unroll
                for (int vr = 0; vr < 8; ++vr) {
                    float t = __shfl(acc[vr], src, 32);
                    if ((lane & 7) == vr) s = t;
                }
                if (lane < 16 && ri < m && (rc & 1ull)) {   // event == 1
                    cnum += slhg[base + ri] - logf(G + s + EPSF);
                    cden += 1.f;
                }
            }
        } else {
            // rare oversize bucket: scalar O(m^2/32) fallback from global (L2-resident)
            for (unsigned i = (unsigned)lane; i < m; i += 32u) {
                ull rc = compKey(skey[base + i], sidxev[base + i]);
                float s = 0.f;
                for (unsigned j = 0; j < m; ++j) {
                    ull cj = compKey(skey[base + j], sidxev[base + j]);
                    if (cj >= rc) s += expf(slhg[base + j]);
                }
                if (rc & 1ull) {
                    cnum += slhg[base + i] - logf(G + s + EPSF);
                    cden += 1.f;
                }
            }
        }
#pragma unroll
        for (int off = 16; off; off >>= 1) {
            cnum += __shfl_xor(cnum, off, 32);
            cden += __shfl_xor(cden, off, 32);
        }
        if (lane == 0) { bnum[bkt] = cnum; bden[bkt] = cden; }
    }
}

// ---- pass G1: deterministic per-stratum reduction of bucket partials ----
__global__ void k_reduceK(const float* __restrict__ bnum, const float* __restrict__ bden,
                          float* numK, float* denK) {
    int k = blockIdx.x;
    __shared__ float pn[256], pd[256];
    int t = threadIdx.x;
    const int CH = NBINS / 256;
    const float* bn = bnum + (size_t)k * NBINS;
    const float* bd = bden + (size_t)k * NBINS;
    float sn = 0.f, sd = 0.f;
    for (int i = 0; i < CH; ++i) { sn += bn[t * CH + i]; sd += bd[t * CH + i]; }
    pn[t] = sn; pd[t] = sd;
    __syncthreads();
    if (t == 0) {
        float an = 0.f, ad = 0.f;
        for (int i = 0; i < 256; ++i) { an += pn[i]; ad += pd[i]; }
        numK[k] = an; denK[k] = ad;
    }
}

// ---- pass G2: final scalar loss ----
__global__ void k_final(const float* __restrict__ numK, const float* __restrict__ denK,
                        float* out) {
    if (blockIdx.x == 0 && threadIdx.x == 0) {
        float loss = 0.f;
        for (int k = 0; k < KSTRATA; ++k) {
            float d = denK[k];
            if (d > 0.f) loss += -numK[k] / fmaxf(d, 1.0f);
        }
        out[0] = loss;
    }
}

extern "C" void kernel_launch(void* const* d_in, const int* in_sizes, int n_in,
                              void* d_out, int out_size, void* d_ws, size_t ws_size,
                              hipStream_t stream) {
    (void)n_in; (void)out_size; (void)ws_size;
    const float* lh   = (const float*)d_in[0];
    const float* dur  = (const float*)d_in[1];
    const float* ev   = (const float*)d_in[2];
    const int*   bidx = (const int*)d_in[3];
    int n = in_sizes[0];

    char* ws = (char*)d_ws;
    size_t off = 0;
    auto alloc = [&](size_t bytes) -> char* {
        char* p = ws + off;
        off += (bytes + 255) & ~(size_t)255;
        return p;
    };
    unsigned* counts  = (unsigned*)alloc((size_t)TOTB * 4);
    unsigned* offsets = (unsigned*)alloc((size_t)TOTB * 4);
    unsigned* cursor  = (unsigned*)alloc((size_t)TOTB * 4);
    float*    bsum    = (float*)   alloc((size_t)TOTB * 4);
    float*    gsuf    = (float*)   alloc((size_t)TOTB * 4);
    float*    bnum    = (float*)   alloc((size_t)TOTB * 4);
    float*    bden    = (float*)   alloc((size_t)TOTB * 4);
    unsigned* gammaU  = (unsigned*)alloc(KSTRATA * 4);
    float*    numK    = (float*)   alloc(KSTRATA * 4);
    float*    denK    = (float*)   alloc(KSTRATA * 4);
    unsigned* skey    = (unsigned*)alloc((size_t)n * 4);
    unsigned* sidxev  = (unsigned*)alloc((size_t)n * 4);
    float*    slhg    = (float*)   alloc((size_t)n * 4);

    k_init   <<<(TOTB + 255) / 256, 256, 0, stream>>>(counts, gammaU);
    k_hist   <<<2048, 256, 0, stream>>>(lh, dur, bidx, counts, gammaU, n);
    k_scanF  <<<1, 1024, 0, stream>>>(counts, offsets, cursor);
    k_scatter<<<(n + 255) / 256, 256, 0, stream>>>(lh, dur, ev, bidx, gammaU, cursor,
                                                   skey, sidxev, slhg, n);
    k_bsum   <<<(TOTB + 255) / 256, 256, 0, stream>>>(counts, offsets, slhg, bsum);
    k_scanR  <<<KSTRATA, 256, 0, stream>>>(bsum, gsuf);
    k_main   <<<8192, PF_BLOCK, 0, stream>>>(counts, offsets, gsuf, skey, sidxev, slhg,
                                             bnum, bden);
    k_reduceK<<<KSTRATA, 256, 0, stream>>>(bnum, bden, numK, denK);
    k_final  <<<1, 32, 0, stream>>>(numK, denK, (float*)d_out);
}